// UtilityLoss_14431090115064
// MI455X (gfx1250) — compile-verified
//
#include <hip/hip_runtime.h>
#include <hip/hip_bf16.h>

// ---- constants from the reference ----
#define NDAYS    250
#define SCALINGF 12.0f
#define ALPHAF   1.0f

#define NBLK 512      // partial-histogram blocks (must be multiple of 4 for WMMA K-steps)
#define NTHR 256      // 8 wave32 waves per block

typedef __attribute__((ext_vector_type(2))) float v2f;
typedef __attribute__((ext_vector_type(8))) float v8f;

// --------------------------------------------------------------------------
// Pass 1: stream 4N flat elements, bin w*t*sigmoid(12*x) by day into LDS,
// write each block's 256-entry partial histogram row to d_ws.
// Faithful index pairing: flat k -> sample i=k>>2, resp j=k&3 (inputs/targets
// sample-major), weights/date index m = k mod N (tiled).
// --------------------------------------------------------------------------
__global__ __launch_bounds__(NTHR) void seg_partial_kernel(
    const float* __restrict__ inputs,   // [N,5]
    const float* __restrict__ targets,  // [N,5]
    const float* __restrict__ weights,  // [N]
    const int*   __restrict__ date,     // [N]
    float* __restrict__ partials,       // [NBLK,256]
    int N)
{
    __shared__ float ldsPi[256];
    const int t = threadIdx.x;
    ldsPi[t] = 0.0f;
    __syncthreads();

    const int total  = 4 * N;
    const int stride = NBLK * NTHR;
    int k = blockIdx.x * NTHR + t;
    int m = k % N;                       // only division; then maintained incrementally

    for (; k < total; k += stride) {
        const int i = k >> 2;
        const int j = k & 3;

        // prefetch next iteration's streaming lines (global_prefetch_b8)
        __builtin_prefetch(inputs  + (((k + stride) >> 2) * 5), 0, 0);
        __builtin_prefetch(targets + (((k + stride) >> 2) * 5), 0, 0);

        const float in = inputs[i * 5 + j];
        const float tg = targets[i * 5 + j];
        const float w  = weights[m];
        const int   d  = date[m] & 255;  // days are 0..249; mask keeps LDS in range

        const float x = 1.0f / (1.0f + __expf(-SCALINGF * in));
        atomicAdd(&ldsPi[d], w * tg * x);   // ds_add_f32 (no return)

        m += stride;
        if (m >= N) m -= N;
    }

    __syncthreads();
    partials[blockIdx.x * 256 + t] = ldsPi[t];   // unconditional: every slot written
}

// --------------------------------------------------------------------------
// Pass 2 (1 block, 8 waves): column-sum the NBLK x 256 partials with
// V_WMMA_F32_16X16X4_F32 (A = ones 16x4  =>  D[m][n] = sum_k B[k][n]),
// then reduce sum(Pi) and sum(Pi^2) and emit the scalar loss.
// --------------------------------------------------------------------------
__global__ __launch_bounds__(NTHR) void reduce_loss_kernel(
    const float* __restrict__ partials, float* __restrict__ out)
{
    __shared__ float Pi[256];
    __shared__ float red[256];
    __shared__ float s1sh;

    const int t    = threadIdx.x;
    const int lane = t & 31;
    const int wave = t >> 5;

    const v2f ones = {1.0f, 1.0f};
    const int col  = (lane & 15);
    const int half = lane >> 4;

    // 16 groups of 16 day-columns; 8 waves -> 2 groups each
    for (int g = wave; g < 16; g += 8) {
        v8f acc = {};
        const int c = g * 16 + col;
        for (int b = 0; b < NBLK; b += 4) {
            const int r = b + 2 * half;            // this lane-half's two K rows
            v2f bm;
            bm[0] = partials[r * 256 + c];
            bm[1] = partials[(r + 1) * 256 + c];
            acc = __builtin_amdgcn_wmma_f32_16x16x4_f32(
                false, ones, false, bm, (short)0, acc, false, false);
        }
        // D row M=0 lives in acc[0] of lanes 0..15 (N = lane); all rows equal.
        if (lane < 16) Pi[g * 16 + lane] = acc[0];
    }
    __syncthreads();

    const float v = Pi[t];   // columns >= NDAYS are exactly zero

    red[t] = v;
    __syncthreads();
    for (int s = 128; s > 0; s >>= 1) { if (t < s) red[t] += red[t + s]; __syncthreads(); }
    if (t == 0) s1sh = red[0];
    __syncthreads();

    red[t] = v * v;
    __syncthreads();
    for (int s = 128; s > 0; s >>= 1) { if (t < s) red[t] += red[t + s]; __syncthreads(); }

    if (t == 0) {
        const float s1 = s1sh;
        const float s2 = red[0];
        out[0] = -ALPHAF * s1 * fmaxf(s1, 0.0f) / s2 / (float)NDAYS;
    }
}

// --------------------------------------------------------------------------
extern "C" void kernel_launch(void* const* d_in, const int* in_sizes, int n_in,
                              void* d_out, int out_size, void* d_ws, size_t ws_size,
                              hipStream_t stream)
{
    const float* inputs  = (const float*)d_in[0];
    const float* targets = (const float*)d_in[1];
    const float* weights = (const float*)d_in[2];
    const int*   date    = (const int*)d_in[3];
    const int N = in_sizes[2];                 // weights length == N samples

    float* partials = (float*)d_ws;            // NBLK * 256 floats = 512 KB

    seg_partial_kernel<<<NBLK, NTHR, 0, stream>>>(inputs, targets, weights, date,
                                                  partials, N);
    reduce_loss_kernel<<<1, NTHR, 0, stream>>>(partials, (float*)d_out);
}